// LeNet_78546361909483
// MI455X (gfx1250) — compile-verified
//
#include <hip/hip_runtime.h>
#include <hip/hip_bf16.h>

// ---------------------------------------------------------------------------
// Mesh-CNN forward for MI455X (gfx1250, wave32).
//  - spmm's are fixed-fanout gathers (3/7/6 nnz per output row) -> no atomics
//  - fc1 (1024x1680 @ 1680x512 fp32) mapped to V_WMMA_F32_16X16X4_F32,
//    register-blocked 2x2 wave tile (32x32 C per wave) for 2x FLOP/byte
//  - fc2 + log_softmax: one wave per row, wave32 shfl_xor reduction
// ---------------------------------------------------------------------------

typedef float v2f __attribute__((ext_vector_type(2)));
typedef float v8f __attribute__((ext_vector_type(8)));

#define BATCH 1024

// ---------------------------------------------------------------------------
// grad kernel: gew/gns per (b, c, f).
//   gf[e]  = sum_{j<3} gv[(e*nf+f)*3+j] * h[b,c, gcols[(e*nf+f)*3+j]]
//   gew    = sum_e gf[e]*ew[f,e] ; gns likewise.
// Stored interleaved {gew,gns} for float2 gathers later.
// ---------------------------------------------------------------------------
__global__ void mesh_grad_kernel(const float* __restrict__ h,
                                 const int* __restrict__ gcols,
                                 const float* __restrict__ gv,
                                 const float* __restrict__ ew,
                                 const float* __restrict__ ns,
                                 float* __restrict__ grad,   // (B*C*nf) x 2
                                 int C, int nv, int nf)
{
    long long tid = (long long)blockIdx.x * blockDim.x + threadIdx.x;
    long long total = (long long)BATCH * C * nf;
    if (tid >= total) return;
    int f = (int)(tid % nf);
    long long bc = tid / nf;                 // b*C + c
    const float* hc = h + bc * nv;
    float gew = 0.f, gns = 0.f;
#pragma unroll
    for (int e = 0; e < 3; ++e) {
        int r = e * nf + f;
        float gf = 0.f;
#pragma unroll
        for (int j = 0; j < 3; ++j) {
            int nz = r * 3 + j;
            gf += gv[nz] * hc[gcols[nz]];
        }
        gew += gf * ew[f * 3 + e];
        gns += gf * ns[f * 3 + e];
    }
    grad[tid * 2 + 0] = gew;
    grad[tid * 2 + 1] = gns;
}

// ---------------------------------------------------------------------------
// conv kernel: one block (64 threads) per (b, v).
// Phase 1: threads t<Cin build feat[t] = {id, lap, gv_ew, gv_ns} in LDS.
// Phase 2: threads t<Cout contract with c[o,i,k], add bias, relu.
// ---------------------------------------------------------------------------
__global__ void mesh_conv_kernel(const float* __restrict__ h,      // (B,Cin,nv)
                                 const float* __restrict__ grad,   // (B,Cin,nf,2)
                                 const int* __restrict__ lcols,
                                 const float* __restrict__ lv,
                                 const int* __restrict__ fcols,
                                 const float* __restrict__ fv,
                                 const float* __restrict__ cw,     // (Cout,Cin,4)
                                 const float* __restrict__ cb,     // (Cout)
                                 float* __restrict__ out,          // (B,Cout,nvp)
                                 int Cin, int Cout, int nv, int nvp, int nf)
{
    __shared__ float feat[40][4];
    int v = (int)(blockIdx.x % nvp);
    int b = (int)(blockIdx.x / nvp);
    int t = threadIdx.x;

    if (t < Cin) {
        const float*  hc = h + ((long long)b * Cin + t) * nv;
        const float2* gc = (const float2*)grad + ((long long)b * Cin + t) * nf;
        float id  = hc[v];
        float lap = 0.f;
#pragma unroll
        for (int k = 0; k < 7; ++k) {
            int nz = v * 7 + k;
            lap += lv[nz] * hc[lcols[nz]];
        }
        float gew = 0.f, gns = 0.f;
#pragma unroll
        for (int k = 0; k < 6; ++k) {
            int nz = v * 6 + k;
            float2 g = gc[fcols[nz]];
            float  w = fv[nz];
            gew += w * g.x;
            gns += w * g.y;
        }
        feat[t][0] = id; feat[t][1] = lap; feat[t][2] = gew; feat[t][3] = gns;
    }
    __syncthreads();
    if (t < Cout) {
        float acc = cb[t];
        const float* w = cw + (long long)t * Cin * 4;
        for (int i = 0; i < Cin; ++i) {
            acc += w[i * 4 + 0] * feat[i][0]
                 + w[i * 4 + 1] * feat[i][1]
                 + w[i * 4 + 2] * feat[i][2]
                 + w[i * 4 + 3] * feat[i][3];
        }
        out[((long long)b * Cout + t) * nvp + v] = fmaxf(acc, 0.f);
    }
}

// ---------------------------------------------------------------------------
// fc1 GEMM via V_WMMA_F32_16X16X4_F32, register-blocked.
//   Y[m,n] = relu( sum_k X[m,k]*W[n,k] + bias[n] ),  M=1024, N=512, K=1680.
// Wave tile = 32x64? No: 2 M-frags x 2 N-frags = 32x32 C per wave,
// 4 accumulators; per K-step: 2 A-frag + 2 B-frag loads feed 4 WMMAs
// (8 B/lane per WMMA vs 16 before). Block = 8 waves (4 M x 2 N) -> 128x64.
// A frag (16x4 f32): lane%16 -> M row, half=lane>>4 selects K pair {0,1}/{2,3}.
// B frag (4x16 f32): lane%16 -> N, same K-pair addressing (W row-major = B^T).
// C frag: vgpr r -> M = half*8 + r, N = lane%16.
// ---------------------------------------------------------------------------
__global__ void fc1_wmma_kernel(const float* __restrict__ X,    // (1024,1680)
                                const float* __restrict__ W,    // (512,1680)
                                const float* __restrict__ bias, // (512)
                                float* __restrict__ Y)          // (1024,512)
{
    const int K = 1680, N = 512;
    int lane = threadIdx.x & 31;
    int wave = threadIdx.x >> 5;
    int wm = wave & 3;            // 4 wave-tiles along M
    int wn = wave >> 2;           // 2 wave-tiles along N
    int m0 = blockIdx.y * 128 + wm * 32;
    int n0 = blockIdx.x * 64  + wn * 32;
    int half = lane >> 4;
    int l16  = lane & 15;

    const float* arow0 = X + (long long)(m0 + l16) * K + half * 2;
    const float* arow1 = arow0 + 16 * (long long)K;
    const float* brow0 = W + (long long)(n0 + l16) * K + half * 2;
    const float* brow1 = brow0 + 16 * (long long)K;

    v8f acc00 = {}, acc01 = {}, acc10 = {}, acc11 = {};
#pragma unroll 2
    for (int k = 0; k < K; k += 4) {
        v2f a0 = *(const v2f*)(arow0 + k);
        v2f a1 = *(const v2f*)(arow1 + k);
        v2f b0 = *(const v2f*)(brow0 + k);
        v2f b1 = *(const v2f*)(brow1 + k);
        acc00 = __builtin_amdgcn_wmma_f32_16x16x4_f32(false, a0, false, b0,
                                                      (short)0, acc00, false, false);
        acc01 = __builtin_amdgcn_wmma_f32_16x16x4_f32(false, a0, false, b1,
                                                      (short)0, acc01, false, false);
        acc10 = __builtin_amdgcn_wmma_f32_16x16x4_f32(false, a1, false, b0,
                                                      (short)0, acc10, false, false);
        acc11 = __builtin_amdgcn_wmma_f32_16x16x4_f32(false, a1, false, b1,
                                                      (short)0, acc11, false, false);
    }

    v8f accs[2][2] = { { acc00, acc01 }, { acc10, acc11 } };
#pragma unroll
    for (int j = 0; j < 2; ++j) {
        int n = n0 + j * 16 + l16;
        float bn = bias[n];
#pragma unroll
        for (int i = 0; i < 2; ++i) {
#pragma unroll
            for (int r = 0; r < 8; ++r) {
                int m = m0 + i * 16 + half * 8 + r;
                Y[(long long)m * N + n] = fmaxf(accs[i][j][r] + bn, 0.f);
            }
        }
    }
}

// ---------------------------------------------------------------------------
// fc2 + log_softmax: one wave (32 lanes) per batch row.
// ---------------------------------------------------------------------------
__global__ void fc2_logsoftmax_kernel(const float* __restrict__ A,   // (1024,512)
                                      const float* __restrict__ W,   // (10,512)
                                      const float* __restrict__ bias,// (10)
                                      float* __restrict__ out)       // (1024,10)
{
    int b = blockIdx.x;
    int lane = threadIdx.x;
    const float* a = A + (long long)b * 512;
    float logit[10];
#pragma unroll
    for (int n = 0; n < 10; ++n) {
        float p = 0.f;
        for (int k = lane; k < 512; k += 32)
            p += a[k] * W[n * 512 + k];
#pragma unroll
        for (int off = 16; off > 0; off >>= 1)
            p += __shfl_xor(p, off, 32);
        logit[n] = p + bias[n];
    }
    float mx = logit[0];
#pragma unroll
    for (int n = 1; n < 10; ++n) mx = fmaxf(mx, logit[n]);
    float s = 0.f;
#pragma unroll
    for (int n = 0; n < 10; ++n) s += __expf(logit[n] - mx);
    float lse = __logf(s) + mx;
    if (lane < 10)
        out[(long long)b * 10 + lane] = logit[lane] - lse;
}

// ---------------------------------------------------------------------------
// Host side
// ---------------------------------------------------------------------------
static int find_input(const int* sizes, int n, int want, int occ)
{
    int c = 0;
    for (int i = 0; i < n; ++i)
        if (sizes[i] == want) { if (c == occ) return i; ++c; }
    return 0;
}

extern "C" void kernel_launch(void* const* d_in, const int* in_sizes, int n_in,
                              void* d_out, int out_size, void* d_ws, size_t ws_size,
                              hipStream_t stream)
{
    // Level geometry
    static const int NF[3]  = {5120, 1280, 320};
    static const int NVP[3] = {642, 162, 42};
    static const int CIN[3]  = {1, 20, 40};
    static const int COUT[3] = {20, 40, 40};

    // Resolve inputs by unique flat size (robust to dict-flattening order).
    // Collisions: ew{i}/ns{i} (occ 0/1) and b2/b3 (occ 0/1) — order-stable
    // under both insertion-order and sorted-key flattening.
    const float* x = (const float*)d_in[find_input(in_sizes, n_in, BATCH * 2562, 0)];

    const int*   gidx[3]; const int* lidx[3]; const int* fidx[3];
    const float* gv[3]; const float* lvv[3]; const float* fvv[3];
    const float* ew[3]; const float* ns[3];
    const float* cw[3]; const float* cb[3];
    int bocc = 0;
    for (int i = 0; i < 3; ++i) {
        int nf = NF[i], nvp = NVP[i];
        gidx[i] = (const int*)d_in[find_input(in_sizes, n_in, 2 * 9 * nf, 0)];
        lidx[i] = (const int*)d_in[find_input(in_sizes, n_in, 2 * 7 * nvp, 0)];
        fidx[i] = (const int*)d_in[find_input(in_sizes, n_in, 2 * 6 * nvp, 0)];
        gv[i]  = (const float*)d_in[find_input(in_sizes, n_in, 9 * nf, 0)];
        lvv[i] = (const float*)d_in[find_input(in_sizes, n_in, 7 * nvp, 0)];
        fvv[i] = (const float*)d_in[find_input(in_sizes, n_in, 6 * nvp, 0)];
        ew[i]  = (const float*)d_in[find_input(in_sizes, n_in, 3 * nf, 0)];
        ns[i]  = (const float*)d_in[find_input(in_sizes, n_in, 3 * nf, 1)];
        cw[i]  = (const float*)d_in[find_input(in_sizes, n_in, COUT[i] * CIN[i] * 4, 0)];
        int bsz = COUT[i];
        int occ = (bsz == 40) ? bocc++ : 0;   // b2 (occ0) / b3 (occ1) share size 40
        cb[i]  = (const float*)d_in[find_input(in_sizes, n_in, bsz, occ)];
    }
    const float* fc1_w = (const float*)d_in[find_input(in_sizes, n_in, 512 * 1680, 0)];
    const float* fc1_b = (const float*)d_in[find_input(in_sizes, n_in, 512, 0)];
    const float* fc2_w = (const float*)d_in[find_input(in_sizes, n_in, 10 * 512, 0)];
    const float* fc2_b = (const float*)d_in[find_input(in_sizes, n_in, 10, 0)];

    // Workspace layout (floats)
    float* ws = (float*)d_ws;
    float* h1 = ws;                                         // 1024*20*642
    float* h2 = h1 + (size_t)BATCH * 20 * 642;              // 1024*40*162
    float* h3 = h2 + (size_t)BATCH * 40 * 162;              // 1024*40*42 = (1024,1680)
    float* a1 = h3 + (size_t)BATCH * 40 * 42;               // 1024*512
    float* grad = a1 + (size_t)BATCH * 512;                 // max 1024*20*1280*2

    const float* hin[3] = { x, h1, h2 };
    float* hout[3] = { h1, h2, h3 };

    for (int i = 0; i < 3; ++i) {
        int C = CIN[i], Co = COUT[i], nv = (i == 0 ? 2562 : NVP[i - 1]);
        int nf = NF[i], nvp = NVP[i];
        const int* gcols = gidx[i] + 9 * nf;   // skip rows half of stacked (2,N)
        const int* lcols = lidx[i] + 7 * nvp;
        const int* fcols = fidx[i] + 6 * nvp;

        long long gtotal = (long long)BATCH * C * nf;
        int gblocks = (int)((gtotal + 255) / 256);
        hipLaunchKernelGGL(mesh_grad_kernel, dim3(gblocks), dim3(256), 0, stream,
                           hin[i], gcols, gv[i], ew[i], ns[i], grad, C, nv, nf);

        int cblocks = BATCH * nvp;
        hipLaunchKernelGGL(mesh_conv_kernel, dim3(cblocks), dim3(64), 0, stream,
                           hin[i], grad, lcols, lvv[i], fcols, fvv[i],
                           cw[i], cb[i], hout[i], C, Co, nv, nvp, nf);
    }

    // fc1: (1024,1680) @ (1680,512) with f32 WMMA; 128x64 block tile,
    // 32x32 register-blocked wave tile (4 accumulators).
    hipLaunchKernelGGL(fc1_wmma_kernel, dim3(512 / 64, 1024 / 128), dim3(256), 0, stream,
                       h3, fc1_w, fc1_b, a1);

    // fc2 + log_softmax
    hipLaunchKernelGGL(fc2_logsoftmax_kernel, dim3(BATCH), dim3(32), 0, stream,
                       a1, fc2_w, fc2_b, (float*)d_out);
}